// GPT2Attention_12043088298297
// MI455X (gfx1250) — compile-verified
//
#include <hip/hip_runtime.h>
#include <math.h>

#define HIDDEN    1024
#define NUM_HEADS 16
#define HEAD_DIM  64
#define BATCH     32
#define T_CACHED  2047
#define T_TOTAL   2048
#define QK_SCALE  0.125f   // HEAD_DIM^-0.5

typedef __attribute__((ext_vector_type(2))) float v2f;
typedef __attribute__((ext_vector_type(8))) float v8f;

// ---------------------------------------------------------------------------
// Generic f32 GEMM: C[M,N] = A[M,K] @ B[K,N] + bias[N], via V_WMMA_F32_16X16X4_F32
// One wave (32 threads) per 16x16 output tile. M,N multiples of 16, K multiple of 4.
// f32 A-frag layout: lanes 0-15 hold {K=0,K=1}, lanes 16-31 hold {K=2,K=3}.
// f32 B-frag layout: VGPR j holds row K=j (lanes 0-15) / row K=j+2 (lanes 16-31).
// f32 C/D layout:    VGPR r holds row M=r (lanes 0-15) / row M=r+8 (lanes 16-31).
// ---------------------------------------------------------------------------
__global__ __launch_bounds__(32)
void gemm_wmma_f32(const float* __restrict__ A, const float* __restrict__ Bm,
                   const float* __restrict__ bias, float* __restrict__ C,
                   int M, int N, int K) {
  const int lane = threadIdx.x & 31;
  const int half = lane >> 4;   // 0: lanes 0-15, 1: lanes 16-31
  const int l16  = lane & 15;
  const int tn = blockIdx.x * 16;
  const int tm = blockIdx.y * 16;

  v8f acc = {};
  const float* arow = A + (size_t)(tm + l16) * K;
  for (int k0 = 0; k0 < K; k0 += 4) {
    const int ka = k0 + 2 * half;
    const float2 av = *(const float2*)(arow + ka);   // contiguous pair -> b64 load
    v2f a, b;
    a[0] = av.x;
    a[1] = av.y;
    b[0] = Bm[(size_t)(ka + 0) * N + tn + l16];
    b[1] = Bm[(size_t)(ka + 1) * N + tn + l16];
    acc = __builtin_amdgcn_wmma_f32_16x16x4_f32(
        /*neg_a=*/false, a, /*neg_b=*/false, b,
        /*c_mod=*/(short)0, acc, /*reuse_a=*/false, /*reuse_b=*/false);
  }

  const float bv = bias[tn + l16];
  const int rbase = tm + half * 8;
#pragma unroll
  for (int r = 0; r < 8; ++r) {
    C[(size_t)(rbase + r) * N + tn + l16] = acc[r] + bv;
  }
}

// ---------------------------------------------------------------------------
// Fused single-query attention + K/V concat materialization.
// One workgroup (256 threads = 8 waves) per (batch, head).
//  Phase 0: load q, write new-token k/v into outK/outV[t=2047], score[2047]
//  Phase 1: stream K-cache rows (float4), copy to outK, dot with q -> scores
//  Phase 2: block-wide max / exp / sum (softmax) in LDS
//  Phase 3: stream V-cache rows (float4), copy to outV, accumulate p·V
// ---------------------------------------------------------------------------
__global__ __launch_bounds__(256)
void decode_attn_fused(const float* __restrict__ key_cache,
                       const float* __restrict__ value_cache,
                       const int*   __restrict__ seq_lens,
                       const float* __restrict__ qkv,      // [B, 3*HIDDEN]
                       float* __restrict__ outK,           // [B,NH,T_TOTAL,D]
                       float* __restrict__ outV,           // [B,NH,T_TOTAL,D]
                       float* __restrict__ attn_ws) {      // [B, HIDDEN]
  __shared__ float  s_q[HEAD_DIM];
  __shared__ float  s_vnew[HEAD_DIM];
  __shared__ float  s_scores[T_TOTAL];
  __shared__ float  s_red[256];
  __shared__ float4 s_acc[256];

  const int bh  = blockIdx.x;
  const int b   = bh / NUM_HEADS;
  const int h   = bh % NUM_HEADS;
  const int tid = threadIdx.x;
  const int slen = seq_lens[b];

  const float* qrow = qkv + (size_t)b * 3 * HIDDEN + h * HEAD_DIM;
  const float* krow = qrow + HIDDEN;
  const float* vrow = qrow + 2 * HIDDEN;

  float*       outKbh = outK + (size_t)bh * T_TOTAL * HEAD_DIM;
  float*       outVbh = outV + (size_t)bh * T_TOTAL * HEAD_DIM;
  const float* kcbh   = key_cache   + (size_t)bh * T_CACHED * HEAD_DIM;
  const float* vcbh   = value_cache + (size_t)bh * T_CACHED * HEAD_DIM;

  // ---- Phase 0: new token ----
  if (tid < HEAD_DIM) {
    const float qv = qrow[tid];
    const float kn = krow[tid];
    const float vn = vrow[tid];
    s_q[tid]    = qv;
    s_vnew[tid] = vn;
    outKbh[(size_t)T_CACHED * HEAD_DIM + tid] = kn;
    outVbh[(size_t)T_CACHED * HEAD_DIM + tid] = vn;
    s_red[tid] = qv * kn;
  }
  __syncthreads();
  if (tid == 0) {
    float dot = 0.f;
#pragma unroll
    for (int i = 0; i < HEAD_DIM; ++i) dot += s_red[i];
    s_scores[T_CACHED] = dot * QK_SCALE;   // freshly appended token: always visible
  }
  __syncthreads();

  // ---- Phase 1: scores for cached rows + K copy ----
  const int j = tid & 15;   // 1/16-row chunk (4 floats)
  const int g = tid >> 4;   // row group 0..15
  const float4 q4 = *(const float4*)(s_q + j * 4);
  for (int t0 = 0; t0 < T_CACHED; t0 += 16) {
    const int t = t0 + g;
    float p = 0.f;
    if (t < T_CACHED) {
      const size_t off = (size_t)t * HEAD_DIM + j * 4;
      __builtin_prefetch(kcbh + off + 16 * HEAD_DIM, 0, 0);
      const float4 kv = *(const float4*)(kcbh + off);
      *(float4*)(outKbh + off) = kv;
      p = q4.x * kv.x + q4.y * kv.y + q4.z * kv.z + q4.w * kv.w;
    }
    // reduce across the 16-lane half-wave group
    p += __shfl_xor(p, 1);
    p += __shfl_xor(p, 2);
    p += __shfl_xor(p, 4);
    p += __shfl_xor(p, 8);
    if (j == 0 && t < T_CACHED) {
      s_scores[t] = (t < slen) ? p * QK_SCALE : -1e30f;
    }
  }
  __syncthreads();

  // ---- Phase 2: softmax (max, exp, sum) ----
  float m = -1e30f;
  for (int t = tid; t < T_TOTAL; t += 256) m = fmaxf(m, s_scores[t]);
  s_red[tid] = m;
  __syncthreads();
  for (int off = 128; off > 0; off >>= 1) {
    if (tid < off) s_red[tid] = fmaxf(s_red[tid], s_red[tid + off]);
    __syncthreads();
  }
  m = s_red[0];
  __syncthreads();

  float sum = 0.f;
  for (int t = tid; t < T_TOTAL; t += 256) {
    const float e = __expf(s_scores[t] - m);
    s_scores[t] = e;
    sum += e;
  }
  s_red[tid] = sum;
  __syncthreads();
  for (int off = 128; off > 0; off >>= 1) {
    if (tid < off) s_red[tid] += s_red[tid + off];
    __syncthreads();
  }
  const float inv = 1.0f / s_red[0];
  __syncthreads();

  // ---- Phase 3: p·V accumulation + V copy ----
  float4 acc = {0.f, 0.f, 0.f, 0.f};
  for (int t = g; t < T_CACHED; t += 16) {
    const size_t off = (size_t)t * HEAD_DIM + j * 4;
    __builtin_prefetch(vcbh + off + 16 * HEAD_DIM, 0, 0);
    const float4 v = *(const float4*)(vcbh + off);
    *(float4*)(outVbh + off) = v;
    const float p = s_scores[t];
    acc.x += p * v.x; acc.y += p * v.y; acc.z += p * v.z; acc.w += p * v.w;
  }
  if (g == (T_CACHED & 15)) {   // row 2047 handled by its residue group
    const float p = s_scores[T_CACHED];
    acc.x += p * s_vnew[j * 4 + 0];
    acc.y += p * s_vnew[j * 4 + 1];
    acc.z += p * s_vnew[j * 4 + 2];
    acc.w += p * s_vnew[j * 4 + 3];
  }
  s_acc[tid] = acc;
  __syncthreads();
  if (tid < HEAD_DIM) {
    const int jj = tid >> 2;   // which float4 chunk
    const int c  = tid & 3;    // component
    float tot = 0.f;
#pragma unroll
    for (int rr = 0; rr < 16; ++rr) {
      tot += ((const float*)&s_acc[rr * 16 + jj])[c];
    }
    attn_ws[(size_t)b * HIDDEN + h * HEAD_DIM + tid] = tot * inv;
  }
}

// ---------------------------------------------------------------------------
extern "C" void kernel_launch(void* const* d_in, const int* in_sizes, int n_in,
                              void* d_out, int out_size, void* d_ws, size_t ws_size,
                              hipStream_t stream) {
  const float* hidden_states = (const float*)d_in[0];  // [B,1,H]
  const float* key_cache     = (const float*)d_in[1];  // [B,NH,T_CACHED,D]
  const float* value_cache   = (const float*)d_in[2];  // [B,NH,T_CACHED,D]
  // d_in[3]: block_tables (unused by reference)
  const int*   seq_lens      = (const int*)d_in[4];    // [B]
  // d_in[5]: max_seq_len (scalar, unused)
  const float* W_attn        = (const float*)d_in[6];  // [H, 3H]
  const float* b_attn        = (const float*)d_in[7];  // [3H]
  const float* W_proj        = (const float*)d_in[8];  // [H, H]
  const float* b_proj        = (const float*)d_in[9];  // [H]

  float* out      = (float*)d_out;
  float* attn_out = out;                                      // [B,1,H] = 32768
  float* outK     = out + (size_t)BATCH * HIDDEN;             // [B,NH,T_TOTAL,D]
  float* outV     = outK + (size_t)BATCH * NUM_HEADS * T_TOTAL * HEAD_DIM;

  float* qkv_ws  = (float*)d_ws;                              // [B, 3H]
  float* attn_ws = qkv_ws + (size_t)BATCH * 3 * HIDDEN;       // [B, H]

  // 1) QKV projection: [32,1024] @ [1024,3072] + b_attn
  {
    dim3 grid(3 * HIDDEN / 16, BATCH / 16);
    gemm_wmma_f32<<<grid, 32, 0, stream>>>(hidden_states, W_attn, b_attn, qkv_ws,
                                           BATCH, 3 * HIDDEN, HIDDEN);
  }

  // 2) Fused attention + K/V concat materialization
  {
    dim3 grid(BATCH * NUM_HEADS);
    decode_attn_fused<<<grid, 256, 0, stream>>>(key_cache, value_cache, seq_lens,
                                                qkv_ws, outK, outV, attn_ws);
  }

  // 3) Output projection: [32,1024] @ [1024,1024] + b_proj
  {
    dim3 grid(HIDDEN / 16, BATCH / 16);
    gemm_wmma_f32<<<grid, 32, 0, stream>>>(attn_ws, W_proj, b_proj, attn_out,
                                           BATCH, HIDDEN, HIDDEN);
  }
}